// Post_processing_11278584119621
// MI455X (gfx1250) — compile-verified
//
#include <hip/hip_runtime.h>
#include <math.h>

#define NCLS 2
#define NTOT 1344
#define CONF_THRF 0.7f
#define IOU_THRF 0.2f

typedef __attribute__((ext_vector_type(2))) float v2f;
typedef __attribute__((ext_vector_type(8))) float v8f;

#if defined(__gfx1250__) && defined(__has_builtin)
#if __has_builtin(__builtin_amdgcn_wmma_f32_16x16x4_f32)
#define USE_WMMA_F32 1
#endif
#endif

// ---------------------------------------------------------------------------
// Kernel A: decode levels, assemble 18-dim offsets, conf mask
// ---------------------------------------------------------------------------
__global__ void kdecode(const float* __restrict__ out3, const float* __restrict__ out4,
                        const float* __restrict__ out5,
                        const float* __restrict__ o13, const float* __restrict__ o14,
                        const float* __restrict__ o15,
                        const float* __restrict__ of3, const float* __restrict__ of4,
                        const float* __restrict__ of5,
                        float* __restrict__ conf, float* __restrict__ cls,
                        float* __restrict__ pos, float* __restrict__ offf,
                        int* __restrict__ mask)
{
    int r = blockIdx.x * blockDim.x + threadIdx.x;
    if (r >= NTOT) return;
    const float *po, *p1, *pf; int S, st, idx;
    if (r < 1024)      { po = out3; p1 = o13; pf = of3; S = 32; st = 4;  idx = r; }
    else if (r < 1280) { po = out4; p1 = o14; pf = of4; S = 16; st = 8;  idx = r - 1024; }
    else               { po = out5; p1 = o15; pf = of5; S = 8;  st = 16; idx = r - 1280; }
    int HW = S * S;
    int ii = idx / S, jj = idx - ii * S;

    float cf = po[idx];
    float c0 = po[HW + idx];
    float c1 = po[2 * HW + idx];
    float x  = po[3 * HW + idx] * (float)st + (float)(st * jj);
    float y  = po[4 * HW + idx] * (float)st + (float)(st * ii);

    float o[18];
    float minx =  INFINITY, maxx = -INFINITY;
    float miny =  INFINITY, maxy = -INFINITY;
    #pragma unroll
    for (int k = 0; k < 18; k++) {
        float v = pf[k * HW + idx] + p1[k * HW + idx] + ((k & 1) ? x : y);
        o[k] = v;
        if (k & 1) { minx = fminf(minx, v); maxx = fmaxf(maxx, v); }
        else       { miny = fminf(miny, v); maxy = fmaxf(maxy, v); }
    }
    float area = (maxx - minx) * (maxy - miny) / 40.0f;
    float sg = 1.0f / (1.0f + expf(-area));
    mask[r] = (cf > CONF_THRF * sg) ? 1 : 0;
    conf[r] = cf;
    cls[2 * r] = c0; cls[2 * r + 1] = c1;
    pos[2 * r] = x;  pos[2 * r + 1] = y;
    #pragma unroll
    for (int k = 0; k < 18; k++) offf[r * 18 + k] = o[k];
}

// ---------------------------------------------------------------------------
// Kernel B: min-area polygon via V_WMMA_F32_16X16X4_F32 (one wave per box),
// then box decode + rotated corners.
// U = P(9x2) @ [c;s](2x16)   V = P @ [-s;c]  per 16-rotation chunk.
// ---------------------------------------------------------------------------
__global__ void __launch_bounds__(32) kpoly(const float* __restrict__ offf,
                                            float* __restrict__ rect,
                                            float* __restrict__ corners,
                                            float* __restrict__ areas5)
{
    int box  = blockIdx.x;
    int lane = threadIdx.x;
    __shared__ float sP[18];
    __shared__ float sU[16][16];
    __shared__ float sV[16][16];
    __shared__ float s_cs[72];
    __shared__ float s_st[36][4];

    if (lane < 18) sP[lane] = offf[box * 18 + lane];
    __syncthreads();

    int l16 = lane & 15;

    // B operand: column q holds point q -> (a_q, b_q) = (comp0, comp1).
    // f32 B 4x16 layout: lane N (0-15) holds K0 in v0, K1 in v1; lanes 16-31 hold K2/K3 (zeroed).
    v2f bv;
    {
        int q = l16;
        bool bok = (lane < 16) && (q < 9);
        bv.x = bok ? sP[2 * q]     : 0.0f;
        bv.y = bok ? sP[2 * q + 1] : 0.0f;
    }

    for (int t = 0; t < 3; t++) {
        int p = 16 * t + l16;
        bool pok = (lane < 16) && (p < 36);
        float c = 0.0f, s = 0.0f;
        if (pok) {
            // triu_indices(9, k=1), row-major
            int rem = p, i = 0, cnt = 8;
            while (rem >= cnt) { rem -= cnt; cnt--; i++; }
            int j = i + 1 + rem;
            float d0 = sP[2 * j]     - sP[2 * i];
            float d1 = sP[2 * j + 1] - sP[2 * i + 1];
            float nr = sqrtf(d0 * d0 + d1 * d1 + 1e-12f);
            c = d0 / nr; s = d1 / nr;
        }
#ifdef USE_WMMA_F32
        // A 16x4 layout: lane M holds K0/K1 in v0/v1; lanes 16-31 hold K2/K3 (zero).
        v2f au; au.x = c;  au.y = s;   // row p of D_u = c*a + s*b
        v2f av; av.x = -s; av.y = c;   // row p of D_v = -s*a + c*b
        v8f z = {0.f,0.f,0.f,0.f,0.f,0.f,0.f,0.f};
        v8f du = __builtin_amdgcn_wmma_f32_16x16x4_f32(false, au, false, bv, (short)0, z, false, false);
        v8f dv = __builtin_amdgcn_wmma_f32_16x16x4_f32(false, av, false, bv, (short)0, z, false, false);
        {
            // C/D layout: lanes 0-15 -> (M=r, N=lane); lanes 16-31 -> (M=8+r, N=lane-16)
            int col = lane & 15, half = lane >> 4;
            #pragma unroll
            for (int r2 = 0; r2 < 8; r2++) {
                sU[half * 8 + r2][col] = du[r2];
                sV[half * 8 + r2][col] = dv[r2];
            }
        }
#else
        if (lane < 16) {
            for (int q = 0; q < 16; q++) {
                float a = (q < 9) ? sP[2 * q]     : 0.0f;
                float b = (q < 9) ? sP[2 * q + 1] : 0.0f;
                sU[l16][q] =  a * c + b * s;
                sV[l16][q] = -a * s + b * c;
            }
        }
#endif
        __syncthreads();
        if (pok) {
            float umin = sU[l16][0], umax = umin;
            float vmin = sV[l16][0], vmax = vmin;
            #pragma unroll
            for (int q = 1; q < 9; q++) {
                umin = fminf(umin, sU[l16][q]); umax = fmaxf(umax, sU[l16][q]);
                vmin = fminf(vmin, sV[l16][q]); vmax = fmaxf(vmax, sV[l16][q]);
            }
            s_st[p][0] = umin; s_st[p][1] = umax;
            s_st[p][2] = vmin; s_st[p][3] = vmax;
            s_cs[2 * p] = c; s_cs[2 * p + 1] = s;
        }
        __syncthreads();
    }

    if (lane == 0) {
        float best = INFINITY; int k = 0;
        for (int p = 0; p < 36; p++) {
            float ar = (s_st[p][1] - s_st[p][0]) * (s_st[p][3] - s_st[p][2]);
            if (ar < best) { best = ar; k = p; }   // first-occurrence argmin
        }
        float c = s_cs[2 * k], s = s_cs[2 * k + 1];
        float umin = s_st[k][0], umax = s_st[k][1];
        float vmin = s_st[k][2], vmax = s_st[k][3];
        float cu[4] = { umin, umax, umax, umin };
        float cv[4] = { vmin, vmin, vmax, vmax };
        float rl[8];
        #pragma unroll
        for (int q = 0; q < 4; q++) {
            rl[2 * q]     = cu[q] * c - cv[q] * s;
            rl[2 * q + 1] = cu[q] * s + cv[q] * c;
        }
        #pragma unroll
        for (int q = 0; q < 8; q++) rect[box * 8 + q] = rl[q];

        // _box_decoder + _rbox_corners
        float w = sqrtf((rl[0]-rl[2])*(rl[0]-rl[2]) + (rl[1]-rl[3])*(rl[1]-rl[3]));
        float h = sqrtf((rl[4]-rl[2])*(rl[4]-rl[2]) + (rl[5]-rl[3])*(rl[5]-rl[3]));
        float cx = (rl[1] + rl[3] + rl[5] + rl[7]) * 0.25f;
        float cy = (rl[0] + rl[2] + rl[4] + rl[6]) * 0.25f;
        float ang = atanf((rl[0] - rl[2]) / (rl[1] - rl[3]));
        float cc = cosf(ang), sn = sinf(ang);
        const float lx[4] = { -0.5f, 0.5f, 0.5f, -0.5f };
        const float ly[4] = { -0.5f, -0.5f, 0.5f, 0.5f };
        #pragma unroll
        for (int q = 0; q < 4; q++) {
            corners[box * 8 + 2 * q]     = cx + lx[q] * w * cc - ly[q] * h * sn;
            corners[box * 8 + 2 * q + 1] = cy + lx[q] * w * sn + ly[q] * h * cc;
        }
        areas5[box] = w * h;
    }
}

// ---------------------------------------------------------------------------
// Kernel C: exact rank == argsort(key)[::-1] with stable-sort tie semantics
// ---------------------------------------------------------------------------
__global__ void korder(const float* __restrict__ conf, const int* __restrict__ mask,
                       int* __restrict__ order)
{
    int i = blockIdx.x * blockDim.x + threadIdx.x;
    if (i >= NTOT) return;
    float ki = mask[i] ? conf[i] : -INFINITY;
    int rank = 0;
    for (int j = 0; j < NTOT; j++) {
        float kj = mask[j] ? conf[j] : -INFINITY;
        if (kj > ki || (kj == ki && j > i)) rank++;
    }
    order[rank] = i;
}

// ---------------------------------------------------------------------------
// NMS helpers
// ---------------------------------------------------------------------------
__device__ __forceinline__ bool pin_quad(float p0, float p1, const float* q)
{
    bool ge = true, le = true;
    #pragma unroll
    for (int k = 0; k < 4; k++) {
        int kn = (k + 1) & 3;
        float e0 = q[2 * kn]     - q[2 * k];
        float e1 = q[2 * kn + 1] - q[2 * k + 1];
        float r0 = p0 - q[2 * k];
        float r1 = p1 - q[2 * k + 1];
        float cr = e0 * r1 - e1 * r0;
        ge = ge && (cr >= 0.0f);
        le = le && (cr <= 0.0f);
    }
    return ge || le;
}

__device__ float quad_inter(const float* A, const float* B)
{
    float rA0[4], rA1[4], rB0[4], rB1[4];
    #pragma unroll
    for (int k = 0; k < 4; k++) {
        int kn = (k + 1) & 3;
        rA0[k] = A[2 * kn]     - A[2 * k];
        rA1[k] = A[2 * kn + 1] - A[2 * k + 1];
        rB0[k] = B[2 * kn]     - B[2 * k];
        rB1[k] = B[2 * kn + 1] - B[2 * k + 1];
    }
    float px[24], py[24];
    int   vl[24];
    #pragma unroll
    for (int k = 0; k < 4; k++) {
        px[k] = A[2 * k]; py[k] = A[2 * k + 1];
        vl[k] = pin_quad(A[2 * k], A[2 * k + 1], B) ? 1 : 0;      // inB
        px[4 + k] = B[2 * k]; py[4 + k] = B[2 * k + 1];
        vl[4 + k] = pin_quad(B[2 * k], B[2 * k + 1], A) ? 1 : 0;  // inA
    }
    for (int a = 0; a < 4; a++) {
        for (int b = 0; b < 4; b++) {
            float qp0 = B[2 * b]     - A[2 * a];
            float qp1 = B[2 * b + 1] - A[2 * a + 1];
            float den = rA0[a] * rB1[b] - rA1[a] * rB0[b];
            float ds  = (fabsf(den) < 1e-9f) ? 1e-9f : den;
            float t   = (qp0 * rB1[b] - qp1 * rB0[b]) / ds;
            float uu  = (qp0 * rA1[a] - qp1 * rA0[a]) / ds;
            int ok = (fabsf(den) > 1e-9f) && (t >= 0.0f) && (t <= 1.0f)
                                         && (uu >= 0.0f) && (uu <= 1.0f);
            int m = 8 + 4 * a + b;
            px[m] = A[2 * a]     + t * rA0[a];
            py[m] = A[2 * a + 1] + t * rA1[a];
            vl[m] = ok;
        }
    }
    int n = 0; float sx = 0.0f, sy = 0.0f;
    for (int m = 0; m < 24; m++) if (vl[m]) { n++; sx += px[m]; sy += py[m]; }
    float inv = 1.0f / (float)((n > 1) ? n : 1);
    float cx = sx * inv, cy = sy * inv;
    float ang[24];
    for (int m = 0; m < 24; m++) {
        px[m] -= cx; py[m] -= cy;
        ang[m] = vl[m] ? atan2f(py[m], px[m]) : 1e9f;
    }
    // stable insertion sort by angle (ascending) == jnp.argsort stable
    for (int m = 1; m < 24; m++) {
        float ka = ang[m], kx = px[m], ky = py[m]; int kv = vl[m];
        int q2 = m - 1;
        while (q2 >= 0 && ang[q2] > ka) {
            ang[q2 + 1] = ang[q2]; px[q2 + 1] = px[q2];
            py[q2 + 1] = py[q2];   vl[q2 + 1] = vl[q2];
            q2--;
        }
        ang[q2 + 1] = ka; px[q2 + 1] = kx; py[q2 + 1] = ky; vl[q2 + 1] = kv;
    }
    float r0x = px[0], r0y = py[0];
    for (int m = 0; m < 24; m++) if (!vl[m]) { px[m] = r0x; py[m] = r0y; }
    float sacc = 0.0f;
    for (int m = 0; m < 24; m++) {
        int mn = (m + 1) % 24;
        sacc += px[m] * py[mn] - py[m] * px[mn];
    }
    float area = 0.5f * fabsf(sacc);
    return (n >= 3) ? area : 0.0f;
}

// ---------------------------------------------------------------------------
// Kernel D: sequential NMS scan (single workgroup, suppression map in LDS)
// ---------------------------------------------------------------------------
__global__ void __launch_bounds__(1024) knms(const int* __restrict__ order,
                                             const int* __restrict__ mask,
                                             const float* __restrict__ pos,
                                             const float* __restrict__ rect,
                                             const float* __restrict__ corners,
                                             const float* __restrict__ areas5,
                                             int* __restrict__ keep)
{
    __shared__ int   sup[NTOT];
    __shared__ int   sact;
    __shared__ float sQ[8], sC[8], sA;
    int tid = threadIdx.x, bd = blockDim.x;
    for (int j = tid; j < NTOT; j += bd) { sup[j] = mask[j] ? 0 : 1; keep[j] = 0; }
    __syncthreads();

    for (int it = 0; it < NTOT; it++) {
        if (tid == 0) {
            int i   = order[it];
            int act = sup[i] ? 0 : 1;
            keep[i] = act;
            sact = act;
            if (act) {
                for (int q = 0; q < 8; q++) { sQ[q] = rect[8 * i + q]; sC[q] = corners[8 * i + q]; }
                sA = areas5[i];
            }
        }
        __syncthreads();
        if (sact) {
            for (int j = tid; j < NTOT; j += bd) {
                if (!sup[j]) {
                    int jn = j + bd; if (jn >= NTOT) jn = j;
                    __builtin_prefetch(&corners[8 * jn], 0, 1);   // global_prefetch path
                    float p0 = pos[2 * j + 1];  // y
                    float p1 = pos[2 * j];      // x
                    bool supj = pin_quad(p0, p1, sQ);             // inside quad i
                    if (!supj) {
                        float inter = quad_inter(sC, &corners[8 * j]);
                        float iou = inter / (sA + areas5[j] - inter + 1e-9f);
                        supj = (iou >= IOU_THRF);
                    }
                    if (supj) sup[j] = 1;
                }
            }
        }
        __syncthreads();
    }
}

// ---------------------------------------------------------------------------
// Kernel E: result + rect_out
// ---------------------------------------------------------------------------
__global__ void kout(const int* __restrict__ keep, const float* __restrict__ cls,
                     const float* __restrict__ pos, const float* __restrict__ rect,
                     float* __restrict__ out)
{
    int j = blockIdx.x * blockDim.x + threadIdx.x;
    if (j >= NTOT) return;
    float c0 = cls[2 * j], c1 = cls[2 * j + 1];
    float mx = fmaxf(c0, c1);
    int kp = keep[j];
    int m0 = kp && (c0 == mx);
    int m1 = kp && (c1 == mx);
    float x = pos[2 * j], y = pos[2 * j + 1];
    out[2 + (0 * NTOT + j) * 2 + 0] = m0 ? x : 0.0f;
    out[2 + (0 * NTOT + j) * 2 + 1] = m0 ? y : 0.0f;
    out[2 + (1 * NTOT + j) * 2 + 0] = m1 ? x : 0.0f;
    out[2 + (1 * NTOT + j) * 2 + 1] = m1 ? y : 0.0f;
    float kf = kp ? 1.0f : 0.0f;
    #pragma unroll
    for (int t = 0; t < 8; t++)
        out[2 + 2 * NTOT * 2 + j * 8 + t] = rect[8 * j + t] * kf;
}

// ---------------------------------------------------------------------------
// Kernel F: cnt reduction (2 classes)
// ---------------------------------------------------------------------------
__global__ void __launch_bounds__(256) kcnt(const int* __restrict__ keep,
                                            const float* __restrict__ cls,
                                            float* __restrict__ out)
{
    __shared__ float r0[256], r1[256];
    int tid = threadIdx.x;
    float a0 = 0.0f, a1 = 0.0f;
    for (int j = tid; j < NTOT; j += 256) {
        float c0 = cls[2 * j], c1 = cls[2 * j + 1];
        float mx = fmaxf(c0, c1);
        if (keep[j]) {
            if (c0 == mx) a0 += 1.0f;
            if (c1 == mx) a1 += 1.0f;
        }
    }
    r0[tid] = a0; r1[tid] = a1;
    __syncthreads();
    for (int s2 = 128; s2 > 0; s2 >>= 1) {
        if (tid < s2) { r0[tid] += r0[tid + s2]; r1[tid] += r1[tid + s2]; }
        __syncthreads();
    }
    if (tid == 0) { out[0] = r0[0]; out[1] = r1[0]; }
}

// ---------------------------------------------------------------------------
extern "C" void kernel_launch(void* const* d_in, const int* in_sizes, int n_in,
                              void* d_out, int out_size, void* d_ws, size_t ws_size,
                              hipStream_t stream)
{
    (void)in_sizes; (void)n_in; (void)out_size; (void)ws_size;
    const float* out3 = (const float*)d_in[0];
    const float* out4 = (const float*)d_in[1];
    const float* out5 = (const float*)d_in[2];
    const float* o13  = (const float*)d_in[3];
    const float* o14  = (const float*)d_in[4];
    const float* o15  = (const float*)d_in[5];
    const float* of3  = (const float*)d_in[6];
    const float* of4  = (const float*)d_in[7];
    const float* of5  = (const float*)d_in[8];

    float* f       = (float*)d_ws;
    float* conf    = f;
    float* cls     = f + 1 * NTOT;
    float* pos     = f + 3 * NTOT;
    float* offf    = f + 5 * NTOT;
    float* rect    = f + 23 * NTOT;
    float* corners = f + 31 * NTOT;
    float* areas5  = f + 39 * NTOT;
    int*   ip      = (int*)(f + 40 * NTOT);
    int*   mask    = ip;
    int*   order   = ip + 1 * NTOT;
    int*   keep    = ip + 2 * NTOT;

    dim3 g((NTOT + 255) / 256), b(256);
    kdecode<<<g, b, 0, stream>>>(out3, out4, out5, o13, o14, o15, of3, of4, of5,
                                 conf, cls, pos, offf, mask);
    kpoly<<<NTOT, 32, 0, stream>>>(offf, rect, corners, areas5);
    korder<<<g, b, 0, stream>>>(conf, mask, order);
    knms<<<1, 1024, 0, stream>>>(order, mask, pos, rect, corners, areas5, keep);
    kout<<<g, b, 0, stream>>>(keep, cls, pos, rect, (float*)d_out);
    kcnt<<<1, 256, 0, stream>>>(keep, cls, (float*)d_out);
}